// Multiheaded_AAMsoftmax_8607114461544
// MI455X (gfx1250) — compile-verified
//
#include <hip/hip_runtime.h>
#include <math.h>

// ---------------- problem constants ----------------
#define B_   128
#define H_   8
#define DK   64
#define C_   50000
#define NBLK 782            // ceil((C_/16)/4): 64 columns per block, 16 per wave
#define LDA  68             // LDS row stride for A (pad 64 -> 68: 16B aligned, no bank conflicts)

#define S_SC   30.0f
#define COS_M_ 0.98006657784124163f
#define SIN_M_ 0.19866933079506122f
#define TH_C   (-0.98006657784124163f)   // cos(pi - 0.2)
#define MM_C   0.03973386615901224f      // sin(pi - 0.2) * 0.2

typedef __attribute__((ext_vector_type(2))) float v2f;
typedef __attribute__((ext_vector_type(8))) float v8f;

#if defined(__has_builtin)
#if __has_builtin(__builtin_amdgcn_wmma_f32_16x16x4_f32)
#define HAVE_WMMA_F32X4 1
#endif
#endif

__device__ __forceinline__ v8f wmma_f32_16x16x4(v2f a, v2f b, v8f c) {
#ifdef HAVE_WMMA_F32X4
  return __builtin_amdgcn_wmma_f32_16x16x4_f32(false, a, false, b, (short)0, c,
                                               false, false);
#else
  // compile-keepalive fallback (wrong math, flags wmma=0 in histogram)
  c[0] += a.x * b.x;
  return c;
#endif
}

__device__ __forceinline__ void lse_merge(float& m, float& s, float m2, float s2) {
  if (m2 > m) { float t = m; m = m2; m2 = t; t = s; s = s2; s2 = t; }
  if (s2 > 0.f) s += s2 * __expf(m2 - m) * 0.f + s2 * expf(m2 - m); // keep exact expf
}

// ---------------- pass 0: normalize x rows, zero loss ----------------
__global__ __launch_bounds__(256) void k_norm_x(const float* __restrict__ x,
                                                float* __restrict__ xn,
                                                float* __restrict__ loss_slot) {
  if (blockIdx.x == 0 && threadIdx.x == 0) *loss_slot = 0.f;
  const int wid  = blockIdx.x * 8 + (threadIdx.x >> 5);   // 1024 waves = B_*H_
  const int lane = threadIdx.x & 31;
  const int b = wid >> 3, h = wid & 7;
  const float v0 = x[b * 512 + h * 64 + lane];
  const float v1 = x[b * 512 + h * 64 + 32 + lane];
  float ss = v0 * v0 + v1 * v1;
  #pragma unroll
  for (int m = 16; m >= 1; m >>= 1) ss += __shfl_xor(ss, m, 32);
  const float sc = 1.0f / fmaxf(sqrtf(ss), 1e-12f);
  xn[(h * B_ + b) * DK + lane]      = v0 * sc;
  xn[(h * B_ + b) * DK + 32 + lane] = v1 * sc;
}

// ---------------- pass 1: WMMA GEMM + margin + online softmax stats ----------------
__global__ __launch_bounds__(128) void k_gemm_stats(
    const float* __restrict__ w, const float* __restrict__ xn,
    const int* __restrict__ label, float2* __restrict__ stats,
    float* __restrict__ lablogit) {
  __shared__ float  lsA[B_ * LDA];
  __shared__ int    lsLab[B_];
  __shared__ float2 lsStat[4][B_];

  const int h    = blockIdx.y;
  const int tid  = threadIdx.x;
  const int wv   = tid >> 5;
  const int lane = tid & 31;
  const int half = lane >> 4;
  const int l15  = lane & 15;

  { // cooperative stage: A (this head) -> LDS, labels -> LDS
    const float4* src = (const float4*)(xn + (size_t)(h * B_ + tid) * DK);
    float4* dst = (float4*)(lsA + tid * LDA);
    #pragma unroll
    for (int i = 0; i < 16; ++i) dst[i] = src[i];
    lsLab[tid] = label[tid];
  }
  __syncthreads();

  const int  col0  = blockIdx.x * 64 + wv * 16;
  const int  colL  = col0 + l15;
  const bool valid = (col0 < C_);

  if (valid) {
    v8f acc[8];
    #pragma unroll
    for (int mt = 0; mt < 8; ++mt)
      acc[mt] = (v8f){0.f, 0.f, 0.f, 0.f, 0.f, 0.f, 0.f, 0.f};
    float wss = 0.f;
    const float* wbase = w + (size_t)h * DK * C_ + colL;
    #pragma unroll
    for (int k0 = 0; k0 < DK; k0 += 4) {
      const int kA = k0 + half * 2;
      v2f bf;
      bf.x = wbase[(size_t)(kA + 0) * C_];
      bf.y = wbase[(size_t)(kA + 1) * C_];
      wss += bf.x * bf.x + bf.y * bf.y;          // weight column sumsq (inline norm)
      #pragma unroll
      for (int mt = 0; mt < 8; ++mt) {
        const float* ap = lsA + (mt * 16 + l15) * LDA + kA;
        v2f af; af.x = ap[0]; af.y = ap[1];
        acc[mt] = wmma_f32_16x16x4(af, bf, acc[mt]);
      }
    }
    wss += __shfl_xor(wss, 16, 32);              // fold odd/even K halves
    const float cscale = 1.0f / fmaxf(sqrtf(wss), 1e-12f);

    #pragma unroll
    for (int mt = 0; mt < 8; ++mt) {
      #pragma unroll
      for (int j = 0; j < 8; ++j) {
        const int row = mt * 16 + j + half * 8;
        const float cosv = acc[mt][j] * cscale;
        float logit;
        if (colL == lsLab[row]) {
          const float sine = sqrtf(fmaxf(1.0f - cosv * cosv, 0.0f));
          const float phi  = cosv * COS_M_ - sine * SIN_M_;
          logit = ((cosv - TH_C > 0.f) ? phi : (cosv - MM_C)) * S_SC;
          lablogit[h * B_ + row] = logit;        // unique writer per (h,row)
        } else {
          logit = cosv * S_SC;
        }
        // row-wise (over 16 cols in this half-wave) max + sum-exp
        float m = logit;
        #pragma unroll
        for (int msk = 8; msk >= 1; msk >>= 1) m = fmaxf(m, __shfl_xor(m, msk, 32));
        float e = expf(logit - m);
        #pragma unroll
        for (int msk = 8; msk >= 1; msk >>= 1) e += __shfl_xor(e, msk, 32);
        if (l15 == 0) lsStat[wv][row] = make_float2(m, e);
      }
    }
  } else if (l15 == 0) {
    #pragma unroll
    for (int mt = 0; mt < 8; ++mt)
      #pragma unroll
      for (int j = 0; j < 8; ++j)
        lsStat[wv][mt * 16 + j + half * 8] = make_float2(-__builtin_inff(), 0.f);
  }
  __syncthreads();

  if (tid < B_) {  // merge the 4 waves' tiles -> one (m,s) per row per block
    float m = -__builtin_inff(), s = 0.f;
    #pragma unroll
    for (int ww = 0; ww < 4; ++ww) {
      const float2 p = lsStat[ww][tid];
      lse_merge(m, s, p.x, p.y);
    }
    stats[(size_t)(h * B_ + tid) * NBLK + blockIdx.x] = make_float2(m, s);
  }
}

// ---------------- pass 2: fold stats -> (rowmax, 1/Z), loss ----------------
__global__ __launch_bounds__(256) void k_reduce(
    const float2* __restrict__ stats, const float* __restrict__ lablogit,
    float* __restrict__ rowmax, float* __restrict__ invz,
    float* __restrict__ loss) {
  __shared__ float sm[256], ss[256];
  const int hb = blockIdx.x;
  float m = -__builtin_inff(), s = 0.f;
  for (int i = threadIdx.x; i < NBLK; i += 256) {
    const float2 p = stats[(size_t)hb * NBLK + i];
    lse_merge(m, s, p.x, p.y);
  }
  sm[threadIdx.x] = m; ss[threadIdx.x] = s;
  __syncthreads();
  for (int st = 128; st > 0; st >>= 1) {
    if (threadIdx.x < st) {
      m = sm[threadIdx.x]; s = ss[threadIdx.x];
      lse_merge(m, s, sm[threadIdx.x + st], ss[threadIdx.x + st]);
      sm[threadIdx.x] = m; ss[threadIdx.x] = s;
    }
    __syncthreads();
  }
  if (threadIdx.x == 0) {
    m = sm[0]; s = ss[0];
    rowmax[hb] = m;
    invz[hb]   = 1.0f / s;
    const float nll = m + logf(s) - lablogit[hb];
    atomicAdd(loss, nll * (1.0f / (B_ * H_)));
  }
}

// ---------------- pass 3: recompute (weight is L2-resident), head-avg softmax ----------------
__global__ __launch_bounds__(128) void k_output(
    const float* __restrict__ w, const float* __restrict__ xn,
    const int* __restrict__ label, const float* __restrict__ rowmax,
    const float* __restrict__ invz, float* __restrict__ out) {
  __shared__ float lsA[B_ * LDA];
  __shared__ int   lsLab[B_];
  __shared__ float lsM[B_], lsZ[B_];

  const int tid  = threadIdx.x;
  const int wv   = tid >> 5;
  const int lane = tid & 31;
  const int half = lane >> 4;
  const int l15  = lane & 15;
  const int col0 = blockIdx.x * 64 + wv * 16;
  const int colL = col0 + l15;
  const bool valid = (col0 < C_);

  lsLab[tid] = label[tid];

  v8f oacc[8];
  #pragma unroll
  for (int mt = 0; mt < 8; ++mt)
    oacc[mt] = (v8f){0.f, 0.f, 0.f, 0.f, 0.f, 0.f, 0.f, 0.f};

  for (int h = 0; h < H_; ++h) {
    __syncthreads();
    { // stage this head's A + row stats
      const float4* src = (const float4*)(xn + (size_t)(h * B_ + tid) * DK);
      float4* dst = (float4*)(lsA + tid * LDA);
      #pragma unroll
      for (int i = 0; i < 16; ++i) dst[i] = src[i];
      lsM[tid] = rowmax[h * B_ + tid];
      lsZ[tid] = invz[h * B_ + tid];
    }
    __syncthreads();
    if (!valid) continue;

    v8f acc[8];
    #pragma unroll
    for (int mt = 0; mt < 8; ++mt)
      acc[mt] = (v8f){0.f, 0.f, 0.f, 0.f, 0.f, 0.f, 0.f, 0.f};
    float wss = 0.f;
    const float* wbase = w + (size_t)h * DK * C_ + colL;
    #pragma unroll
    for (int k0 = 0; k0 < DK; k0 += 4) {
      const int kA = k0 + half * 2;
      v2f bf;
      bf.x = wbase[(size_t)(kA + 0) * C_];
      bf.y = wbase[(size_t)(kA + 1) * C_];
      wss += bf.x * bf.x + bf.y * bf.y;
      #pragma unroll
      for (int mt = 0; mt < 8; ++mt) {
        const float* ap = lsA + (mt * 16 + l15) * LDA + kA;
        v2f af; af.x = ap[0]; af.y = ap[1];
        acc[mt] = wmma_f32_16x16x4(af, bf, acc[mt]);
      }
    }
    wss += __shfl_xor(wss, 16, 32);
    const float cscale = 1.0f / fmaxf(sqrtf(wss), 1e-12f);

    #pragma unroll
    for (int mt = 0; mt < 8; ++mt) {
      #pragma unroll
      for (int j = 0; j < 8; ++j) {
        const int row = mt * 16 + j + half * 8;
        const float cosv = acc[mt][j] * cscale;
        float logit;
        if (colL == lsLab[row]) {
          const float sine = sqrtf(fmaxf(1.0f - cosv * cosv, 0.0f));
          const float phi  = cosv * COS_M_ - sine * SIN_M_;
          logit = ((cosv - TH_C > 0.f) ? phi : (cosv - MM_C)) * S_SC;
        } else {
          logit = cosv * S_SC;
        }
        const float prob = expf(logit - lsM[row]) * lsZ[row];
        oacc[mt][j] += (1.0f / H_) * prob;
      }
    }
  }

  if (valid) {
    #pragma unroll
    for (int mt = 0; mt < 8; ++mt) {
      #pragma unroll
      for (int j = 0; j < 8; ++j) {
        const int row = mt * 16 + j + half * 8;
        out[(size_t)row * C_ + colL] = oacc[mt][j];
      }
    }
  }
}

// ---------------- host launcher ----------------
extern "C" void kernel_launch(void* const* d_in, const int* in_sizes, int n_in,
                              void* d_out, int out_size, void* d_ws, size_t ws_size,
                              hipStream_t stream) {
  const float* x     = (const float*)d_in[0];   // (128, 512) f32
  const float* w     = (const float*)d_in[1];   // (8, 64, 50000) f32
  const int*   label = (const int*)d_in[2];     // (128,) int

  float* out = (float*)d_out;                   // 128*50000 probs + 1 loss
  float* ws  = (float*)d_ws;

  float*  xn       = ws;                        // 65536 f
  float*  lablogit = ws + 65536;                // 1024 f
  float*  rmax     = ws + 66560;                // 1024 f
  float*  iz       = ws + 67584;                // 1024 f
  float2* stats    = (float2*)(ws + 68608);     // 1024*NBLK float2 (~6.4 MB)
  float*  loss     = out + (size_t)B_ * C_;

  k_norm_x   <<<128, 256, 0, stream>>>(x, xn, loss);
  k_gemm_stats<<<dim3(NBLK, H_), 128, 0, stream>>>(w, xn, label, stats, lablogit);
  k_reduce   <<<B_ * H_, 256, 0, stream>>>(stats, lablogit, rmax, iz, loss);
  k_output   <<<NBLK, 128, 0, stream>>>(w, xn, label, rmax, iz, out);
}